// Voxelization_27118423507003
// MI455X (gfx1250) — compile-verified
//
#include <hip/hip_runtime.h>
#include <hip/hip_bf16.h>

#define RDIM 24
#define R3   13824            // 24^3
#define BATCH 32
#define NPTS 131072
#define MEAN_BLOCKS 64        // blocks per batch for mean/max kernels
#define MEAN_BLK 256
#define PTS_PER_THREAD 8      // 131072 / (64*256)
#define SC_WGS_PER_B 8
#define SC_BLK 1024
#define SC_TILES (NPTS / SC_WGS_PER_B / SC_BLK)   // 16

// ws layout (floats):
// [0, 6144)     partial coord sums [B][64][3]
// [6144, 6240)  mean [B][3]
// [6240, 6272)  max norm^2 per batch (float bits, built via uint atomicMax)
#define WS_PARTIAL 0
#define WS_MEAN    6144
#define WS_M2      6240
#define WS_FLOATS  6272

typedef __attribute__((ext_vector_type(2))) float v2f;
typedef __attribute__((ext_vector_type(8))) float v8f;

__global__ void vox_zero_f32(float* __restrict__ p, int n) {
    int i = blockIdx.x * blockDim.x + threadIdx.x;
    if (i < n) p[i] = 0.0f;
}

// ---- pass 1: per-block partial sums of xyz (deterministic tree) ----
__global__ void vox_coord_sum(const float* __restrict__ pts, float* __restrict__ ws) {
    int b   = blockIdx.x / MEAN_BLOCKS;
    int blk = blockIdx.x % MEAN_BLOCKS;
    int tid = threadIdx.x;
    const float* base = pts + ((size_t)b * NPTS + (size_t)blk * (MEAN_BLK * PTS_PER_THREAD)) * 6;
    float ax = 0.f, ay = 0.f, az = 0.f;
    for (int k = 0; k < PTS_PER_THREAD; ++k) {
        const float* p = base + (size_t)(k * MEAN_BLK + tid) * 6;
        ax += p[0]; ay += p[1]; az += p[2];
    }
    __shared__ float sx[MEAN_BLK], sy[MEAN_BLK], sz[MEAN_BLK];
    sx[tid] = ax; sy[tid] = ay; sz[tid] = az;
    __syncthreads();
    for (int s = MEAN_BLK / 2; s > 0; s >>= 1) {
        if (tid < s) { sx[tid] += sx[tid + s]; sy[tid] += sy[tid + s]; sz[tid] += sz[tid + s]; }
        __syncthreads();
    }
    if (tid == 0) {
        float* o = ws + WS_PARTIAL + ((size_t)b * MEAN_BLOCKS + blk) * 3;
        o[0] = sx[0]; o[1] = sy[0]; o[2] = sz[0];
    }
}

// ---- pass 1b: reduce 64 partials -> mean, on the matrix pipe ----
// One wave per batch (EXEC all ones, as WMMA requires).
// D = A(16x4, ones) x B(4x16, partials) + 0  =>  D[0][n] = sum of partials[4n..4n+3].
// B layout (4x16 f32, 2 VGPRs): vgpr0 = {K0 | K2}, vgpr1 = {K1 | K3}, N = lane&15.
__global__ void __launch_bounds__(32) vox_mean_finalize(float* __restrict__ ws) {
    int b = blockIdx.x;
    int lane = threadIdx.x;                   // 0..31, single wave32
    int n  = lane & 15;
    int kh = lane >> 4;                       // 0: K={0,1}, 1: K={2,3}
    int p0 = 4 * n + 2 * kh;                  // partial index for vgpr0
    const float* part = ws + WS_PARTIAL + (size_t)b * MEAN_BLOCKS * 3;

    v2f A; A.x = 1.0f; A.y = 1.0f;            // ones matrix: every slot = 1.0
    const float inv = 1.0f / (float)NPTS;

    for (int c = 0; c < 3; ++c) {
        v2f Bv;
        Bv.x = part[(size_t)p0 * 3 + c];
        Bv.y = part[(size_t)(p0 + 1) * 3 + c];
        v8f Cz = {};
        v8f D = __builtin_amdgcn_wmma_f32_16x16x4_f32(
            /*neg_a=*/false, A, /*neg_b=*/false, Bv,
            /*c_mod=*/(short)0, Cz, /*reuse_a=*/false, /*reuse_b=*/false);
        // D vgpr0: lanes 0..15 hold (M=0, N=lane) = the 16 group sums.
        float t = D[0];
        t += __shfl_down(t, 8, 16);
        t += __shfl_down(t, 4, 16);
        t += __shfl_down(t, 2, 16);
        t += __shfl_down(t, 1, 16);
        if (lane == 0) ws[WS_MEAN + b * 3 + c] = t * inv;
    }
}

// ---- pass 2: max ||p - mean||^2 (uint-bit atomicMax on nonneg floats: deterministic) ----
__global__ void vox_max_norm2(const float* __restrict__ pts, float* __restrict__ ws) {
    int b   = blockIdx.x / MEAN_BLOCKS;
    int blk = blockIdx.x % MEAN_BLOCKS;
    int tid = threadIdx.x;
    float mx = ws[WS_MEAN + b * 3 + 0];
    float my = ws[WS_MEAN + b * 3 + 1];
    float mz = ws[WS_MEAN + b * 3 + 2];
    const float* base = pts + ((size_t)b * NPTS + (size_t)blk * (MEAN_BLK * PTS_PER_THREAD)) * 6;
    float m = 0.f;
    for (int k = 0; k < PTS_PER_THREAD; ++k) {
        const float* p = base + (size_t)(k * MEAN_BLK + tid) * 6;
        float dx = p[0] - mx, dy = p[1] - my, dz = p[2] - mz;
        m = fmaxf(m, dx * dx + dy * dy + dz * dz);
    }
    __shared__ float sm[MEAN_BLK];
    sm[tid] = m;
    __syncthreads();
    for (int s = MEAN_BLK / 2; s > 0; s >>= 1) {
        if (tid < s) sm[tid] = fmaxf(sm[tid], sm[tid + s]);
        __syncthreads();
    }
    if (tid == 0)
        atomicMax((unsigned*)(ws + WS_M2) + b, __float_as_uint(sm[0]));
}

// ---- pass 3: LDS-resident voxel accumulator + async global->LDS staged input ----
__global__ void __launch_bounds__(SC_BLK)
vox_scatter(const float* __restrict__ pts, const float* __restrict__ ws,
            float* __restrict__ out) {
    __shared__ float acc[R3 * 4];            // 216 KB: {count, f1, f2, f3} per voxel
    __shared__ float stage[2][SC_BLK * 6];   // 48 KB: double-buffered point tiles
    int b   = blockIdx.x / SC_WGS_PER_B;
    int w   = blockIdx.x % SC_WGS_PER_B;
    int tid = threadIdx.x;

    for (int i = tid; i < R3 * 4; i += SC_BLK) acc[i] = 0.0f;

    float mx = ws[WS_MEAN + b * 3 + 0];
    float my = ws[WS_MEAN + b * 3 + 1];
    float mz = ws[WS_MEAN + b * 3 + 2];
    float m2 = ws[WS_M2 + b];                              // nonneg float via bit-max
    float denom = (m2 > 0.0f) ? 2.0f * sqrtf(m2) : 1.0f;   // no NaN on degenerate cloud
    __syncthreads();

    const float* src = pts + ((size_t)b * NPTS + (size_t)w * (SC_BLK * SC_TILES)) * 6;
    unsigned ls0 = (unsigned)(size_t)&stage[0][tid * 6];
    unsigned ls1 = (unsigned)(size_t)&stage[1][tid * 6];

    // prime tile 0 -> buffer 0 (inst offset applies to both LDS and global sides)
    {
        const float* g = src + (size_t)tid * 6;
        asm volatile("global_load_async_to_lds_b64 %0, %1, off"            :: "v"(ls0), "v"(g) : "memory");
        asm volatile("global_load_async_to_lds_b64 %0, %1, off offset:8"   :: "v"(ls0), "v"(g) : "memory");
        asm volatile("global_load_async_to_lds_b64 %0, %1, off offset:16"  :: "v"(ls0), "v"(g) : "memory");
    }

    for (int t = 0; t < SC_TILES; ++t) {
        const float* sb = &stage[t & 1][tid * 6];
        if (t + 1 < SC_TILES) {
            unsigned lsn = (t & 1) ? ls0 : ls1;
            const float* g = src + (size_t)((t + 1) * SC_BLK + tid) * 6;
            asm volatile("global_load_async_to_lds_b64 %0, %1, off"           :: "v"(lsn), "v"(g) : "memory");
            asm volatile("global_load_async_to_lds_b64 %0, %1, off offset:8"  :: "v"(lsn), "v"(g) : "memory");
            asm volatile("global_load_async_to_lds_b64 %0, %1, off offset:16" :: "v"(lsn), "v"(g) : "memory");
            asm volatile("s_wait_asynccnt 3" ::: "memory");   // current tile's 3 loads done
        } else {
            asm volatile("s_wait_asynccnt 0" ::: "memory");
        }
        float x = sb[0], y = sb[1], z = sb[2];
        float f1 = sb[3], f2 = sb[4], f3 = sb[5];
        float nx = (x - mx) / denom + 0.5f;
        float ny = (y - my) / denom + 0.5f;
        float nz = (z - mz) / denom + 0.5f;
        float qx = fminf(fmaxf(nx * (float)RDIM, 0.0f), (float)(RDIM - 1));
        float qy = fminf(fmaxf(ny * (float)RDIM, 0.0f), (float)(RDIM - 1));
        float qz = fminf(fmaxf(nz * (float)RDIM, 0.0f), (float)(RDIM - 1));
        int vx = (int)rintf(qx);   // RNE == jnp.round
        int vy = (int)rintf(qy);
        int vz = (int)rintf(qz);
        int idx = (vx * RDIM + vy) * RDIM + vz;
        float* a = acc + idx * 4;
        atomicAdd(a + 0, 1.0f);    // ds_add_f32
        atomicAdd(a + 1, f1);
        atomicAdd(a + 2, f2);
        atomicAdd(a + 3, f3);
    }
    __syncthreads();

    // merge this WG's LDS grid into the global grid (d_out, [B,4,R^3] layout)
    float* ob = out + (size_t)b * 4 * R3;
    for (int i = tid; i < R3; i += SC_BLK) {
        float c0 = acc[i * 4 + 0];
        if (c0 != 0.0f) {
            atomicAdd(ob + 0 * R3 + i, c0);
            atomicAdd(ob + 1 * R3 + i, acc[i * 4 + 1]);
            atomicAdd(ob + 2 * R3 + i, acc[i * 4 + 2]);
            atomicAdd(ob + 3 * R3 + i, acc[i * 4 + 3]);
        }
    }
}

// ---- pass 4: in-place averaging ----
__global__ void vox_finalize(float* __restrict__ out) {
    int g = blockIdx.x * blockDim.x + threadIdx.x;
    if (g >= BATCH * R3) return;
    int b = g / R3, v = g % R3;
    float* ob = out + (size_t)b * 4 * R3;
    float c = ob[v];
    float inv = (c > 0.0f) ? 1.0f / c : 0.0f;
    ob[v] = (c > 0.0f) ? 1.0f : 0.0f;          // ones-channel average
    ob[1 * R3 + v] *= inv;
    ob[2 * R3 + v] *= inv;
    ob[3 * R3 + v] *= inv;
}

extern "C" void kernel_launch(void* const* d_in, const int* in_sizes, int n_in,
                              void* d_out, int out_size, void* d_ws, size_t ws_size,
                              hipStream_t stream) {
    const float* pts = (const float*)d_in[0];
    float* out = (float*)d_out;
    float* ws  = (float*)d_ws;

    int nout = BATCH * 4 * R3;
    vox_zero_f32<<<(nout + 255) / 256, 256, 0, stream>>>(out, nout);
    vox_zero_f32<<<(WS_FLOATS + 255) / 256, 256, 0, stream>>>(ws, WS_FLOATS);
    vox_coord_sum<<<BATCH * MEAN_BLOCKS, MEAN_BLK, 0, stream>>>(pts, ws);
    vox_mean_finalize<<<BATCH, 32, 0, stream>>>(ws);
    vox_max_norm2<<<BATCH * MEAN_BLOCKS, MEAN_BLK, 0, stream>>>(pts, ws);
    vox_scatter<<<BATCH * SC_WGS_PER_B, SC_BLK, 0, stream>>>(pts, ws, out);
    vox_finalize<<<(BATCH * R3 + 255) / 256, 256, 0, stream>>>(out);
}